// NEG_loss_32452772889082
// MI455X (gfx1250) — compile-verified
//
#include <hip/hip_runtime.h>
#include <hip/hip_bf16.h>
#include <math.h>

// ---------------------------------------------------------------------------
// NEG sampled-softmax loss, MI455X (gfx1250, wave32).
// One wave per batch row. The two 16x256 mat-vecs per row are computed with
// V_WMMA_F32_16X16X4_F32 (exact f32), B operand broadcast from LDS.
// Memory-bound: ~285 MB of row gathers -> ~12 us at 23.3 TB/s.
// ---------------------------------------------------------------------------

typedef float v2f __attribute__((ext_vector_type(2)));
typedef float v8f __attribute__((ext_vector_type(8)));

#define BATCH 8192
#define EMBED 256
#define NSAMP 16
#define WAVES_PER_BLOCK 8

__device__ __forceinline__ float log_sigmoid(float x) {
    // numerically stable: min(x,0) - log1p(exp(-|x|))
    return fminf(x, 0.0f) - log1pf(__expf(-fabsf(x)));
}

__global__ void __launch_bounds__(WAVES_PER_BLOCK * 32)
neg_loss_kernel(const int* __restrict__ input_labels,   // (BATCH,2)
                const int* __restrict__ out_labels,     // (BATCH,2)
                const int* __restrict__ noise,          // (BATCH,16)
                const int* __restrict__ cp_noise,       // (BATCH,16)
                const float* __restrict__ in_embed,     // (1e6,256)
                const float* __restrict__ diag_w,       // (4,256)
                float* __restrict__ out)                // scalar
{
    __shared__ __align__(16) float lds_weout[WAVES_PER_BLOCK][EMBED]; // w * e_out
    __shared__ __align__(16) float lds_wein [WAVES_PER_BLOCK][EMBED]; // w_last * e_in

    const int lane = threadIdx.x & 31;
    const int wave = threadIdx.x >> 5;
    const int row  = blockIdx.x * WAVES_PER_BLOCK + wave;
    const int m    = lane & 15;   // sample index for A-matrix rows
    const int half = lane >> 4;   // K-half select per WMMA layout

    const int    t       = input_labels[2 * row + 0];
    const size_t in_idx  = (size_t)input_labels[2 * row + 1];
    const size_t out_idx = (size_t)out_labels[2 * row + 1];

    // --- stage broadcast vectors into LDS; partial target dot -------------
    const int d8 = lane * 8;  // each lane owns 8 contiguous dims
    const float4* wp  = (const float4*)(diag_w + (size_t)t * EMBED + d8);
    const float4* wlp = (const float4*)(diag_w + (size_t)3 * EMBED + d8);
    const float4* eip = (const float4*)(in_embed + in_idx * EMBED + d8);
    const float4* eop = (const float4*)(in_embed + out_idx * EMBED + d8);

    float tp = 0.0f;
#pragma unroll
    for (int q = 0; q < 2; ++q) {
        float4 w = wp[q], wl = wlp[q], ei = eip[q], eo = eop[q];
        float4 weo, wei;
        weo.x = w.x * eo.x;  weo.y = w.y * eo.y;
        weo.z = w.z * eo.z;  weo.w = w.w * eo.w;
        wei.x = wl.x * ei.x; wei.y = wl.y * ei.y;
        wei.z = wl.z * ei.z; wei.w = wl.w * ei.w;
        *(float4*)&lds_weout[wave][d8 + 4 * q] = weo;   // ds_store_b128
        *(float4*)&lds_wein [wave][d8 + 4 * q] = wei;   // ds_store_b128
        tp += weo.x * ei.x + weo.y * ei.y + weo.z * ei.z + weo.w * ei.w;
    }
    // wave-wide reduce of target dot (all 32 lanes active)
#pragma unroll
    for (int mask = 16; mask > 0; mask >>= 1) tp += __shfl_xor(tp, mask, 32);

    // --- gather bases for the 16 noise / 16 cp-noise rows ------------------
    const size_t nu = (size_t)noise   [row * NSAMP + m];
    const size_t nv = (size_t)cp_noise[row * NSAMP + m];
    // A-matrix layout (16x4 f32): lane m, half h holds K = 2h, 2h+1
    const float* pu  = in_embed + nu * EMBED + 2 * half;
    const float* pv  = in_embed + nv * EMBED + 2 * half;
    // B-matrix (4x16 f32): same K map; value independent of column -> LDS bcast
    const float* pbo = &lds_weout[wave][2 * half];
    const float* pbi = &lds_wein [wave][2 * half];

    v8f cu = {0.f, 0.f, 0.f, 0.f, 0.f, 0.f, 0.f, 0.f};
    v8f cv = {0.f, 0.f, 0.f, 0.f, 0.f, 0.f, 0.f, 0.f};

#pragma unroll 4
    for (int d0 = 0; d0 < EMBED; d0 += 4) {
        v2f au = *(const v2f*)(pu  + d0);   // global_load_b64 (gathered row)
        v2f av = *(const v2f*)(pv  + d0);   // global_load_b64
        v2f bu = *(const v2f*)(pbo + d0);   // ds_load_b64 (broadcast)
        v2f bv = *(const v2f*)(pbi + d0);   // ds_load_b64
        // D = A x B + C ; (neg_a, A, neg_b, B, c_mod, C, reuse_a, reuse_b)
        cu = __builtin_amdgcn_wmma_f32_16x16x4_f32(false, au, false, bu,
                                                   (short)0, cu, false, false);
        cv = __builtin_amdgcn_wmma_f32_16x16x4_f32(false, av, false, bv,
                                                   (short)0, cv, false, false);
    }

    // --- epilogue: D[s][n] identical across n; lanes 0 / 16 own s=0..7 / 8..15
    float acc = 0.0f;
    if (m == 0) {
#pragma unroll
        for (int r = 0; r < 8; ++r)
            acc += log_sigmoid(-cu[r]) + log_sigmoid(-cv[r]);
        if (half == 0) acc += 2.0f * log_sigmoid(tp);
    }
#pragma unroll
    for (int mask = 16; mask > 0; mask >>= 1) acc += __shfl_xor(acc, mask, 32);
    if (lane == 0) atomicAdd(out, -acc * (1.0f / (float)BATCH));
}

__global__ void zero_kernel(float* out) { out[0] = 0.0f; }

extern "C" void kernel_launch(void* const* d_in, const int* in_sizes, int n_in,
                              void* d_out, int out_size, void* d_ws, size_t ws_size,
                              hipStream_t stream) {
    (void)in_sizes; (void)n_in; (void)out_size; (void)d_ws; (void)ws_size;
    const int*   input_labels = (const int*)d_in[0];
    const int*   out_labels   = (const int*)d_in[1];
    // d_in[2] = num_sampled scalar (fixed at 16, baked in)
    const int*   noise        = (const int*)d_in[3];
    const int*   cp_noise     = (const int*)d_in[4];
    const float* in_embed     = (const float*)d_in[5];
    const float* diag_w       = (const float*)d_in[6];
    float*       out          = (float*)d_out;

    zero_kernel<<<1, 1, 0, stream>>>(out);
    neg_loss_kernel<<<BATCH / WAVES_PER_BLOCK, WAVES_PER_BLOCK * 32, 0, stream>>>(
        input_labels, out_labels, noise, cp_noise, in_embed, diag_w, out);
}